// JRegularizedCrossEntropyLoss_70841190580811
// MI455X (gfx1250) — compile-verified
//
#include <hip/hip_runtime.h>
#include <hip/hip_bf16.h>

typedef __attribute__((ext_vector_type(2))) float v2f;
typedef __attribute__((ext_vector_type(4))) float v4f;
typedef __attribute__((ext_vector_type(8))) float v8f;

#define C_CLS 8
#define NPIX (512 * 512)
#define BATCH 8
#define NWG 32                            // workgroups per batch
#define WAVES 8                           // 256 threads = 8 wave32
#define PIX_PER_WG (NPIX / NWG)           // 8192
#define PIX_PER_WAVE (PIX_PER_WG / WAVES) // 1024
#define ITERS (PIX_PER_WAVE / 32)         // 32 iterations x 32 pixels
#define TROWS 10                          // 8 data rows + ones row (8) + zeros row (9)
#define TSTRIDE 36                        // row stride (DWORDs): conflict-free + 16B aligned

// d_ws layout (floats): Spart[B*NWG][72] then CEpart[B*NWG]
// Spart row: [0..63] = S[i][k] (class-sums), [64..71] = n[k] (counts)

__global__ __launch_bounds__(256) void jce_main(const float* __restrict__ pred,
                                                const int* __restrict__ target,
                                                float* __restrict__ Spart,
                                                float* __restrict__ CEpart) {
  const int b    = blockIdx.y;
  const int wg   = blockIdx.x;
  const int tid  = threadIdx.x;
  const int lane = tid & 31;
  const int wave = tid >> 5;

  const float* predb = pred + (size_t)b * C_CLS * NPIX;
  const int*   tgtb  = target + (size_t)b * NPIX;

  const int  c8 = lane & 7;   // class this lane loads in phase 1
  const int  g  = lane >> 3;  // pixel group (0..3) in phase 1
  const int  M  = lane & 15;  // WMMA A-row / B-col / D-col index for this lane
  const bool hi = lane >= 16; // upper half-wave: K+2 (A/B), M+8 (D)

  // per-wave staging tile: [row][pixel]; rows 0..7 = classes, 8 = 1.0, 9 = 0.0
  __shared__ float tileSh[WAVES][TROWS][TSTRIDE];
  __shared__ int   ttileSh[WAVES][32];

  // constant rows written once (per-wave region, same-wave LDS is in-order)
  tileSh[wave][8][lane] = 1.f;  // A row M==8 -> accumulates counts n[k]
  tileSh[wave][9][lane] = 0.f;  // A rows M>8 -> zero padding

  v8f  acc = {0.f, 0.f, 0.f, 0.f, 0.f, 0.f, 0.f, 0.f}; // D rows 0..7 = S[i][k], row 8 = n[k]
  float ce_local = 0.f;

  // invariant bases for phase-2 LDS reads (immediate offsets per j)
  const int ridx = (M < 8) ? M : ((M == 8) ? 8 : 9);     // A-source row for this lane
  const float* arow = &tileSh[wave][ridx][hi ? 2 : 0];
  const int*   trow = &ttileSh[wave][hi ? 2 : 0];        // target pairs for one-hot B
  const float* colp = &tileSh[wave][0][lane];            // CE column (stride TSTRIDE)

  const int pstart = wg * PIX_PER_WG + wave * PIX_PER_WAVE;
  const float* gbase = predb + (size_t)c8 * NPIX + 4 * g;

  // ---- software pipeline: prologue load+stage for it=0 ----
  v4f pvA = __builtin_nontemporal_load((const v4f*)(gbase + pstart));
  v4f pvB = __builtin_nontemporal_load((const v4f*)(gbase + pstart + 16));
  int tcur = __builtin_nontemporal_load(tgtb + pstart + lane) & 7;
  *(v4f*)&tileSh[wave][c8][4 * g]      = pvA;
  *(v4f*)&tileSh[wave][c8][16 + 4 * g] = pvB;
  ttileSh[wave][lane] = tcur;

  for (int it = 0; it < ITERS; ++it) {
    // ---- issue next iteration's global loads first (latency hidden by compute) ----
    const bool more = (it + 1) < ITERS;
    const int pnext = pstart + (it + 1) * 32;
    v4f nA = pvA, nB = pvB;
    int tnext = tcur;
    if (more) {
      nA = __builtin_nontemporal_load((const v4f*)(gbase + pnext));
      nB = __builtin_nontemporal_load((const v4f*)(gbase + pnext + 16));
      tnext = __builtin_nontemporal_load(tgtb + pnext + lane) & 7;
    }

    // ---- phase 2: 8x WMMA f32 16x16x4, one-hot B => segmented scatter-add ----
#pragma unroll
    for (int j = 0; j < 8; ++j) {
      const v2f A   = *(const v2f*)&arow[4 * j];      // pred[M][4j + K] (const rows for M>=8)
      const int2 tp = *(const int2*)&trow[4 * j];     // targets of those pixels
      const float bx = (tp.x == M) ? 1.f : 0.f;
      const float by = (tp.y == M) ? 1.f : 0.f;
      const v2f Bm = {bx, by};
      acc = __builtin_amdgcn_wmma_f32_16x16x4_f32(false, A, false, Bm,
                                                  (short)0, acc, false, false);
    }

    // ---- phase 3: cross entropy, one pixel per lane, fully lane-private ----
    float v[C_CLS];
#pragma unroll
    for (int c = 0; c < C_CLS; ++c) v[c] = colp[c * TSTRIDE];
    float m = v[0];
#pragma unroll
    for (int c = 1; c < C_CLS; ++c) m = fmaxf(m, v[c]);
    float s = 0.f;
#pragma unroll
    for (int c = 0; c < C_CLS; ++c) s += __expf(v[c] - m);
    const float lse = m + __logf(s);
    const float vt  = tileSh[wave][tcur][lane]; // picked logit via dynamic LDS read
    ce_local += lse - vt;

    // ---- stage next tile (same-wave LDS in-order: reads above complete first) ----
    if (more) {
      *(v4f*)&tileSh[wave][c8][4 * g]      = nA;
      *(v4f*)&tileSh[wave][c8][16 + 4 * g] = nB;
      ttileSh[wave][lane] = tnext;
      pvA = nA; pvB = nB; tcur = tnext;
    }
  }

  // ---- deterministic in-block reduction (no float atomics) ----
  __shared__ float Sred[WAVES][72];
  __shared__ float ceLds[256];

  // D layout: VGPR r, lanes 0-15 -> (M=r, N=lane); lanes 16-31 -> (M=r+8, N=lane-16)
#pragma unroll
  for (int r = 0; r < 8; ++r) {
    const int Mrow = hi ? (r + 8) : r;
    const int Ncol = M;
    if (Mrow <= 8 && Ncol < 8) Sred[wave][Mrow * 8 + Ncol] = acc[r];
  }
  ceLds[tid] = ce_local;
  __syncthreads();

  const int wgid = b * NWG + wg;
  if (tid < 72) {
    float s = 0.f;
#pragma unroll
    for (int w = 0; w < WAVES; ++w) s += Sred[w][tid];
    Spart[(size_t)wgid * 72 + tid] = s;
  }
  if (tid == 255) {
    float s = 0.f;
    for (int i = 0; i < 256; ++i) s += ceLds[i];
    CEpart[wgid] = s;
  }
}

__global__ __launch_bounds__(640) void jce_final(const float* __restrict__ Spart,
                                                 const float* __restrict__ CEpart,
                                                 const float* __restrict__ wgt,
                                                 float* __restrict__ out) {
  __shared__ float gS[BATCH * 72];
  __shared__ float ceSh;
  const int tid = threadIdx.x;

  if (tid < BATCH * 72) {
    const int b = tid / 72, idx = tid % 72;
    float s = 0.f;
#pragma unroll 1
    for (int w = 0; w < NWG; ++w) s += Spart[(size_t)(b * NWG + w) * 72 + idx];
    gS[tid] = s;
  }
  if (tid == 576) {
    float s = 0.f;
#pragma unroll 1
    for (int i = 0; i < BATCH * NWG; ++i) s += CEpart[i];
    ceSh = s / ((float)BATCH * (float)NPIX);
  }
  __syncthreads();

  if (tid < BATCH) {
    const float* S = &gS[tid * 72];
    const float* n = S + 64;
    float j = 0.f;
#pragma unroll 1
    for (int i = 0; i < C_CLS; ++i) {
      const float diag = S[i * 8 + i] / n[i];
#pragma unroll 1
      for (int k = 0; k < C_CLS; ++k) {
        if (k == i) continue;
        const float aik = S[i * 8 + k] / n[k];
        const float delta = 0.5f * (diag - aik);
        j -= wgt[i * 8 + k] * logf(0.5f + delta);
      }
    }
    out[tid] = j + ceSh;
  }
}

extern "C" void kernel_launch(void* const* d_in, const int* in_sizes, int n_in,
                              void* d_out, int out_size, void* d_ws, size_t ws_size,
                              hipStream_t stream) {
  const float* pred   = (const float*)d_in[0];
  const int*   target = (const int*)d_in[1];
  const float* wgt    = (const float*)d_in[2];
  float* out = (float*)d_out;

  float* Spart  = (float*)d_ws;                       // [B*NWG][72]
  float* CEpart = Spart + (size_t)BATCH * NWG * 72;   // [B*NWG]

  dim3 grid(NWG, BATCH);
  jce_main<<<grid, 256, 0, stream>>>(pred, target, Spart, CEpart);
  jce_final<<<1, 640, 0, stream>>>(Spart, CEpart, wgt, out);
}